// GCN_28638841930048
// MI455X (gfx1250) — compile-verified
//
#include <hip/hip_runtime.h>

// GNN NNConv forward for MI455X (gfx1250, wave32, WMMA f32 16x16x4).
// Edge stage = one GEMM [E, 17*D] @ [17*D, 32]; A rows = outer(e_aug, x_src)
// built in registers.  B (W2aug) is pre-swizzled once per layer into WMMA
// fragment order so the hot loop does ONE global_load_b128 per K-chunk
// (coalesced, L2-resident) instead of 4 strided b32 loads.
// Aggregations: f32 atomics (segment_sum) and uint atomicMax (max-pool on
// relu'd non-negative data, pooled pre-zeroed => empty-graph semantics match).

#define GNN_N 50000
#define GNN_E 100000
#define GNN_H 32
#define GNN_G 2048

typedef float v2f __attribute__((ext_vector_type(2)));
typedef float v8f __attribute__((ext_vector_type(8)));

__device__ __forceinline__ v8f wmma4(v2f a, v2f b, v8f c) {
  // D = A(16x4,f32) * B(4x16,f32) + C(16x16,f32)
  return __builtin_amdgcn_wmma_f32_16x16x4_f32(false, a, false, b, (short)0, c,
                                               false, false);
}

// ---------------------------------------------------------------------------
// K1: y[n,h] = x[n,:] @ root[:,h] + bias[h]      (initializes the aggregator)
// one wave per 16-node tile; N tiles = 3125 (exact)
// ---------------------------------------------------------------------------
template <int D>
__global__ __launch_bounds__(256) void node_root_kernel(
    const float* __restrict__ x, const float* __restrict__ root,
    const float* __restrict__ bias, float* __restrict__ y, int nTiles) {
  int tile = (blockIdx.x * blockDim.x + threadIdx.x) >> 5;
  if (tile >= nTiles) return;
  int lane = threadIdx.x & 31;
  int m = lane & 15;
  bool hi = lane >= 16;
  int n0 = tile * 16;

  const float* arow = x + (n0 + m) * D + (hi ? 2 : 0);
  int cB0 = m + (hi ? 2 * GNN_H : 0);       // B col + lane-half row offset
  int cB1 = 16 + m + (hi ? 2 * GNN_H : 0);

  float bc0 = bias[m], bc1 = bias[16 + m];  // fold bias into C
  v8f acc0, acc1;
#pragma unroll
  for (int j = 0; j < 8; j++) { acc0[j] = bc0; acc1[j] = bc1; }

#pragma unroll
  for (int kk = 0; kk < D; kk += 4) {
    v2f a;  a.x = arow[kk];                 a.y = arow[kk + 1];
    v2f b0; b0.x = root[kk * GNN_H + cB0];  b0.y = root[(kk + 1) * GNN_H + cB0];
    v2f b1; b1.x = root[kk * GNN_H + cB1];  b1.y = root[(kk + 1) * GNN_H + cB1];
    acc0 = wmma4(a, b0, acc0);
    acc1 = wmma4(a, b1, acc1);
  }
#pragma unroll
  for (int j = 0; j < 8; j++) {
    int r = n0 + j + (hi ? 8 : 0);
    y[r * GNN_H + m]      = acc0[j];
    y[r * GNN_H + 16 + m] = acc1[j];
  }
}

// ---------------------------------------------------------------------------
// Pack W2aug into per-(chunk,lane) float4 fragments in exactly the order the
// WMMA B operand wants:  w2p[c*128 + lane*4 + q] = W2aug[r][col]
//   r = 4c + (lane>=16 ? 2:0) + (q&1),  col = (q>>1)*16 + (lane&15)
//   W2aug[(k*D+i), h] = k<16 ? w2[k, i*H+h] : b2[i*H+h]
// ---------------------------------------------------------------------------
__global__ void pack_w2_kernel(const float* __restrict__ w2,
                               const float* __restrict__ b2,
                               float* __restrict__ w2p, int D) {
  int o = blockIdx.x * blockDim.x + threadIdx.x;
  int total = 17 * D * GNN_H;
  if (o >= total) return;
  int c = o >> 7;
  int rem = o & 127;
  int lane = rem >> 2;
  int q = rem & 3;
  int r = c * 4 + ((lane >= 16) ? 2 : 0) + (q & 1);
  int col = ((q >> 1) << 4) + (lane & 15);
  int k = r / D, i = r % D;
  w2p[o] = (k < 16) ? w2[k * (D * GNN_H) + i * GNN_H + col]
                    : b2[i * GNN_H + col];
}

// ---------------------------------------------------------------------------
// K2: per-edge message + scatter-add.
//   e       = relu(edge_attr @ w1 + b1)            [16]
//   msg     = outer([e,1], x_src)  @  W2aug        [32]
//   y[dst] += msg   (atomics)
// ---------------------------------------------------------------------------
template <int D>
__global__ __launch_bounds__(256) void edge_msg_kernel(
    const float* __restrict__ x, const float* __restrict__ ea,
    const float* __restrict__ w1, const float* __restrict__ b1,
    const float* __restrict__ w2p,  // packed W2aug fragments [KO/4][32][4]
    const int* __restrict__ srcI, const int* __restrict__ dstI,
    float* __restrict__ y, int nTiles) {
  int tile = (blockIdx.x * blockDim.x + threadIdx.x) >> 5;
  if (tile >= nTiles) return;
  int lane = threadIdx.x & 31;
  int m = lane & 15;
  bool hi = lane >= 16;
  int e0 = tile * 16;
  int eidx = e0 + m;  // lanes L and L+16 both own edge row m

  // tiny edge MLP (per lane, full row needed locally)
  float a0 = ea[eidx * 4 + 0], a1 = ea[eidx * 4 + 1];
  float a2 = ea[eidx * 4 + 2], a3 = ea[eidx * 4 + 3];
  float eaug[17];
#pragma unroll
  for (int k = 0; k < 16; k++) {
    float v = b1[k] + a0 * w1[k] + a1 * w1[16 + k] + a2 * w1[32 + k] +
              a3 * w1[48 + k];
    eaug[k] = v > 0.f ? v : 0.f;
  }
  eaug[16] = 1.f;  // row carrying b2

  int s = srcI[eidx];
  float xs[D];
#pragma unroll
  for (int i = 0; i < D; i++) xs[i] = x[s * D + i];

  v8f acc0, acc1;
#pragma unroll
  for (int j = 0; j < 8; j++) { acc0[j] = 0.f; acc1[j] = 0.f; }

  const float4* w2p4 = (const float4*)w2p + lane;  // + chunk*32 per step
  constexpr int KO = 17 * D;  // 612 or 544, multiple of 4
#pragma unroll
  for (int kk = 0; kk < KO; kk += 4) {
    // B fragment: one coalesced 16B load per chunk
    float4 f = w2p4[(kk >> 2) * 32];
    // A operand: outer-product elements, lane-half selects K pair
    float axl = eaug[(kk + 0) / D] * xs[(kk + 0) % D];
    float ayl = eaug[(kk + 1) / D] * xs[(kk + 1) % D];
    float axh = eaug[(kk + 2) / D] * xs[(kk + 2) % D];
    float ayh = eaug[(kk + 3) / D] * xs[(kk + 3) % D];
    v2f a;
    a.x = hi ? axh : axl;
    a.y = hi ? ayh : ayl;
    v2f b0;  b0.x = f.x;   b0.y = f.y;
    v2f b1v; b1v.x = f.z;  b1v.y = f.w;
    acc0 = wmma4(a, b0, acc0);
    acc1 = wmma4(a, b1v, acc1);
  }

  // scatter: segment_sum via f32 atomics
#pragma unroll
  for (int j = 0; j < 8; j++) {
    int er = e0 + j + (hi ? 8 : 0);
    int d = dstI[er];
    atomicAdd(&y[d * GNN_H + m], acc0[j]);
    atomicAdd(&y[d * GNN_H + 16 + m], acc1[j]);
  }
}

// ---------------------------------------------------------------------------
__global__ void relu_kernel(float* __restrict__ p, int n) {
  int i = blockIdx.x * blockDim.x + threadIdx.x;
  if (i < n) { float v = p[i]; p[i] = v > 0.f ? v : 0.f; }
}

__global__ void zero_kernel(float* __restrict__ p, int n) {
  int i = blockIdx.x * blockDim.x + threadIdx.x;
  if (i < n) p[i] = 0.f;
}

// max-pool: x >= 0 after relu, pooled pre-zeroed -> uint compare == f32 compare
// and empty graphs stay 0 (matches where(counts>0, pooled, 0)).
__global__ void pool_kernel(const float* __restrict__ x,
                            const int* __restrict__ batch,
                            float* __restrict__ pooled) {
  int idx = blockIdx.x * blockDim.x + threadIdx.x;
  if (idx >= GNN_N * GNN_H) return;
  int n = idx >> 5, h = idx & 31;
  int g = batch[n];
  atomicMax((unsigned int*)&pooled[g * GNN_H + h], __float_as_uint(x[idx]));
}

// head fusion: v = W0@W1 (32), c = b0.W1 + b1
__global__ void head_fuse_kernel(const float* __restrict__ W0,
                                 const float* __restrict__ b0,
                                 const float* __restrict__ W1,
                                 const float* __restrict__ b1,
                                 float* __restrict__ hv) {
  int i = threadIdx.x;
  if (i < GNN_H) {
    float s = 0.f;
    for (int j = 0; j < GNN_H; j++) s += W0[i * GNN_H + j] * W1[j];
    hv[i] = s;
  }
  if (i == 0) {
    float c = b1[0];
    for (int j = 0; j < GNN_H; j++) c += b0[j] * W1[j];
    hv[GNN_H] = c;
  }
}

__global__ void head_out_kernel(const float* __restrict__ pooled,
                                const float* __restrict__ hv,
                                float* __restrict__ out) {
  int g = blockIdx.x * blockDim.x + threadIdx.x;
  if (g >= GNN_G) return;
  float s = hv[GNN_H];
#pragma unroll
  for (int i = 0; i < GNN_H; i++) s += pooled[g * GNN_H + i] * hv[i];
  out[g] = s;
}

// ---------------------------------------------------------------------------
extern "C" void kernel_launch(void* const* d_in, const int* in_sizes, int n_in,
                              void* d_out, int out_size, void* d_ws,
                              size_t ws_size, hipStream_t stream) {
  (void)in_sizes; (void)out_size; (void)ws_size;

  struct Conv { const float *b1, *b2, *bias, *root, *w1, *w2; };
  Conv c[3];
  const float *x, *edge_attr, *lin0W, *lin0b, *lin1W, *lin1b;
  const int *edge_index, *batch;

  if (n_in >= 26) {
    // one d_in pointer per pytree leaf (dict insertion order; params leaves in
    // jax sorted-key order: b1,b2,bias,root,w1,w2 per conv; W,b per lin)
    int p = 0;
    x = (const float*)d_in[p++];
    edge_attr = (const float*)d_in[p++];
    for (int l = 0; l < 3; l++) {
      c[l].b1 = (const float*)d_in[p++];
      c[l].b2 = (const float*)d_in[p++];
      c[l].bias = (const float*)d_in[p++];
      c[l].root = (const float*)d_in[p++];
      c[l].w1 = (const float*)d_in[p++];
      c[l].w2 = (const float*)d_in[p++];
    }
    lin0W = (const float*)d_in[p++]; lin0b = (const float*)d_in[p++];
    lin1W = (const float*)d_in[p++]; lin1b = (const float*)d_in[p++];
    edge_index = (const int*)d_in[p++];
    batch = (const int*)d_in[p++];
  } else {
    // params flattened into a single contiguous blob (same leaf order)
    x = (const float*)d_in[0];
    edge_attr = (const float*)d_in[1];
    const float* pp = (const float*)d_in[2];
    for (int l = 0; l < 3; l++) {
      int D = (l == 0) ? 36 : 32;
      c[l].b1 = pp; pp += 16;
      c[l].b2 = pp; pp += D * GNN_H;
      c[l].bias = pp; pp += GNN_H;
      c[l].root = pp; pp += D * GNN_H;
      c[l].w1 = pp; pp += 4 * 16;
      c[l].w2 = pp; pp += 16 * D * GNN_H;
    }
    lin0W = pp; pp += GNN_H * GNN_H;
    lin0b = pp; pp += GNN_H;
    lin1W = pp; pp += GNN_H;
    lin1b = pp; pp += 1;
    edge_index = (const int*)d_in[3];
    batch = (const int*)d_in[4];
  }
  const int* srcI = edge_index;          // edge_index[0]
  const int* dstI = edge_index + GNN_E;  // edge_index[1]

  float* buf0 = (float*)d_ws;                 // [N,32]
  float* buf1 = buf0 + GNN_N * GNN_H;         // [N,32]
  float* pooled = buf1 + GNN_N * GNN_H;       // [G,32]
  float* headv = pooled + GNN_G * GNN_H;      // [33] (+pad to 48 for align)
  float* w2p = headv + 48;                    // [17*36*32] max, 16B aligned
  float* outp = (float*)d_out;

  const int nodeTiles = GNN_N / 16;  // 3125
  const int edgeTiles = GNN_E / 16;  // 6250
  const int nodeBlks = (nodeTiles * 32 + 255) / 256;  // 391
  const int edgeBlks = (edgeTiles * 32 + 255) / 256;  // 782
  const int nh = GNN_N * GNN_H;

  // ---- layer 0 (D=36): x -> buf0
  node_root_kernel<36><<<nodeBlks, 256, 0, stream>>>(x, c[0].root, c[0].bias,
                                                     buf0, nodeTiles);
  pack_w2_kernel<<<(17 * 36 * GNN_H + 255) / 256, 256, 0, stream>>>(
      c[0].w2, c[0].b2, w2p, 36);
  edge_msg_kernel<36><<<edgeBlks, 256, 0, stream>>>(
      x, edge_attr, c[0].w1, c[0].b1, w2p, srcI, dstI, buf0, edgeTiles);
  relu_kernel<<<(nh + 255) / 256, 256, 0, stream>>>(buf0, nh);

  // ---- layer 1 (D=32): buf0 -> buf1
  node_root_kernel<32><<<nodeBlks, 256, 0, stream>>>(buf0, c[1].root,
                                                     c[1].bias, buf1,
                                                     nodeTiles);
  pack_w2_kernel<<<(17 * 32 * GNN_H + 255) / 256, 256, 0, stream>>>(
      c[1].w2, c[1].b2, w2p, 32);
  edge_msg_kernel<32><<<edgeBlks, 256, 0, stream>>>(
      buf0, edge_attr, c[1].w1, c[1].b1, w2p, srcI, dstI, buf1, edgeTiles);
  relu_kernel<<<(nh + 255) / 256, 256, 0, stream>>>(buf1, nh);

  // ---- layer 2 (D=32): buf1 -> buf0
  node_root_kernel<32><<<nodeBlks, 256, 0, stream>>>(buf1, c[2].root,
                                                     c[2].bias, buf0,
                                                     nodeTiles);
  pack_w2_kernel<<<(17 * 32 * GNN_H + 255) / 256, 256, 0, stream>>>(
      c[2].w2, c[2].b2, w2p, 32);
  edge_msg_kernel<32><<<edgeBlks, 256, 0, stream>>>(
      buf1, edge_attr, c[2].w1, c[2].b1, w2p, srcI, dstI, buf0, edgeTiles);
  relu_kernel<<<(nh + 255) / 256, 256, 0, stream>>>(buf0, nh);

  // ---- global max pool + head
  zero_kernel<<<(GNN_G * GNN_H + 255) / 256, 256, 0, stream>>>(pooled,
                                                               GNN_G * GNN_H);
  pool_kernel<<<(nh + 255) / 256, 256, 0, stream>>>(buf0, batch, pooled);
  head_fuse_kernel<<<1, 32, 0, stream>>>(lin0W, lin0b, lin1W, lin1b, headv);
  head_out_kernel<<<(GNN_G + 255) / 256, 256, 0, stream>>>(pooled, headv,
                                                           outp);
}